// HaarWavelet2D_19241453486095
// MI455X (gfx1250) — compile-verified
//
#include <hip/hip_runtime.h>

// Fused 2-level Haar frequency split for (N,256,256) fp32 images.
// One workgroup per 64x64 output tile; everything intermediate lives in LDS.
// Input tile is brought in with CDNA5 async global->LDS copies (ASYNCcnt);
// outputs are written with non-temporal stores so the 192MB L2 stays
// dedicated to input-halo reuse (outputs are write-once, never re-read).

#define NT   256
#define TILE 64

// LDS plane extents (rows==cols extent, padded row pitch)
#define EX 70   // input tile  (origin i0-3)
#define PX 71
#define EA 69   // LL0 coeffs  (origin i0-3)
#define PA 70
#define EH 65   // |LH|+|HL|+|HH| level 0 (origin i0-1)
#define PH 66
#define EL 68   // low1 = resize255->256(LL0) (origin i0-2)
#define PL 69
#define EP 34   // LL1 / HI1 (origin i0/2-1)
#define PP 35

__device__ __forceinline__ void async_ld_f32(const float* __restrict__ base,
                                             unsigned byteOff, float* ldsPtr) {
    // GVS mode: mem = SGPR64 base + VGPR32 offset; LDS dest addr per lane in VGPR.
    unsigned lds = (unsigned)(unsigned long long)ldsPtr; // low 32 bits = LDS byte addr
    asm volatile("global_load_async_to_lds_b32 %0, %1, %2"
                 :: "v"(lds), "v"(byteOff), "s"(base) : "memory");
}

__device__ __forceinline__ void wait_async0() {
#if defined(__has_builtin) && __has_builtin(__builtin_amdgcn_s_wait_asynccnt)
    __builtin_amdgcn_s_wait_asynccnt(0);
#else
    asm volatile("s_wait_asynccnt 0" ::: "memory");
#endif
}

// jax.image.resize(method='linear'), 255 -> 256, one axis:
// s = (i+0.5)*255/256 - 0.5, clamp to [0,254]; edge renorm == clamped lerp.
__device__ __forceinline__ void map255(int i, int& a0, int& a1, float& f) {
    float s = ((float)i + 0.5f) * (255.0f / 256.0f) - 0.5f;
    s = fminf(fmaxf(s, 0.0f), 254.0f);
    int a = (int)s;
    f  = s - (float)a;
    a0 = a;
    a1 = min(a + 1, 254);
}

// 128 -> 256: s = 0.5*i - 0.25, clamp to [0,127].
__device__ __forceinline__ void map128(int i, int& p0, int& p1, float& f) {
    float s = 0.5f * (float)i - 0.25f;
    s = fminf(fmaxf(s, 0.0f), 127.0f);
    int p = (int)s;
    f  = s - (float)p;
    p0 = p;
    p1 = min(p + 1, 127);
}

__device__ __forceinline__ float bilerp(const float* buf, int pitch,
                                        int r0, int r1, int c0, int c1,
                                        float fr, float fc) {
    float v00 = buf[r0 * pitch + c0], v01 = buf[r0 * pitch + c1];
    float v10 = buf[r1 * pitch + c0], v11 = buf[r1 * pitch + c1];
    float t = v00 + fc * (v01 - v00);
    float b = v10 + fc * (v11 - v10);
    return t + fr * (b - t);
}

__global__ void __launch_bounds__(NT)
haar2_fused_kernel(const float* __restrict__ x,
                   float* __restrict__ out_low,
                   float* __restrict__ out_high)
{
    __shared__ float sX  [EX * PX];
    __shared__ float sLL0[EA * PA];
    __shared__ float sHI0[EH * PH];
    __shared__ float sL1 [EL * PL];
    __shared__ float sLL1[EP * PP];
    __shared__ float sHI1[EP * PP];

    const int tid = threadIdx.x;
    const int j0  = blockIdx.x * TILE;
    const int i0  = blockIdx.y * TILE;
    const int img = blockIdx.z;
    const float* gx = x + (size_t)img * (256 * 256);

    const int oxr = i0 - 3, oxc = j0 - 3;               // sX / sLL0 origin
    const int ohr = i0 - 1, ohc = j0 - 1;               // sHI0 origin
    const int olr = i0 - 2, olc = j0 - 2;               // sL1 origin
    const int opr = (i0 >> 1) - 1, opc = (j0 >> 1) - 1; // sLL1/sHI1 origin

    // ---- Phase 0: async global->LDS copy of the clamped 70x70 input tile ----
    for (int idx = tid; idx < EX * EX; idx += NT) {
        int r = idx / EX, c = idx - r * EX;
        int gr = min(max(oxr + r, 0), 255);
        int gc = min(max(oxc + c, 0), 255);
        unsigned goff = (unsigned)((gr << 8) | gc) << 2;
        async_ld_f32(gx, goff, &sX[r * PX + c]);
    }
    wait_async0();
    __syncthreads();

    // ---- Phase 1: level-0 Haar coeffs (stride 1, overlapping 2x2) ----
    for (int idx = tid; idx < EA * EA; idx += NT) {     // LL0, coeff a = oxr + r
        int r = idx / EA, c = idx - r * EA;             // x local row == r
        float x00 = sX[r * PX + c],       x01 = sX[r * PX + c + 1];
        float x10 = sX[(r + 1) * PX + c], x11 = sX[(r + 1) * PX + c + 1];
        sLL0[r * PA + c] = (((x00 + x01) + x10) + x11) * 0.25f;
    }
    for (int idx = tid; idx < EH * EH; idx += NT) {     // |LH|+|HL|+|HH|, a = ohr + r
        int r = idx / EH, c = idx - r * EH;
        int xr = r + 2, xc = c + 2;                     // (ohr+r) - oxr = r+2
        float x00 = sX[xr * PX + xc],       x01 = sX[xr * PX + xc + 1];
        float x10 = sX[(xr + 1) * PX + xc], x11 = sX[(xr + 1) * PX + xc + 1];
        float lh = (((x00 + x01) - x10) - x11) * 0.25f;
        float hl = (((x00 - x01) + x10) - x11) * 0.25f;
        float hh = (((x00 - x01) - x10) + x11) * 0.25f;
        sHI0[r * PH + c] = (fabsf(lh) + fabsf(hl)) + fabsf(hh);
    }
    __syncthreads();

    // ---- Phase 2: low1 = resize 255->256 of LL0 (tile region + halo) ----
    for (int idx = tid; idx < EL * EL; idx += NT) {
        int r = idx / EL, c = idx - r * EL;
        int gi = olr + r, gj = olc + c;                 // global low1 coords
        int a0, a1, b0, b1; float fa, fb;
        map255(gi, a0, a1, fa);
        map255(gj, b0, b1, fb);
        sL1[r * PL + c] = bilerp(sLL0, PA, a0 - oxr, a1 - oxr,
                                 b0 - oxc, b1 - oxc, fa, fb);
    }
    __syncthreads();

    // ---- Phase 3: level-1 Haar (stride 2 on low1) -> LL1, HI1 ----
    for (int idx = tid; idx < EP * EP; idx += NT) {
        int r = idx / EP, c = idx - r * EP;
        int lr = 2 * r, lc = 2 * c;                     // 2*(opr+r) - olr == 2r
        float y00 = sL1[lr * PL + lc],       y01 = sL1[lr * PL + lc + 1];
        float y10 = sL1[(lr + 1) * PL + lc], y11 = sL1[(lr + 1) * PL + lc + 1];
        float ll = (((y00 + y01) + y10) + y11) * 0.25f;
        float lh = (((y00 + y01) - y10) - y11) * 0.25f;
        float hl = (((y00 - y01) + y10) - y11) * 0.25f;
        float hh = (((y00 - y01) - y10) + y11) * 0.25f;
        sLL1[r * PP + c] = ll;
        sHI1[r * PP + c] = (fabsf(lh) + fabsf(hl)) + fabsf(hh);
    }
    __syncthreads();

    // ---- Phase 4: final resizes + combine; non-temporal write-once outputs ----
    const size_t obase = (size_t)img * (256 * 256);
    for (int idx = tid; idx < TILE * TILE; idx += NT) {
        int r = idx >> 6, c = idx & 63;
        int i = i0 + r, j = j0 + c;

        int p0, p1, q0, q1; float fp, fq;
        map128(i, p0, p1, fp);
        map128(j, q0, q1, fq);
        int lp0 = p0 - opr, lp1 = p1 - opr, lq0 = q0 - opc, lq1 = q1 - opc;
        float lowv = bilerp(sLL1, PP, lp0, lp1, lq0, lq1, fp, fq);
        float hi1v = bilerp(sHI1, PP, lp0, lp1, lq0, lq1, fp, fq);

        int a0, a1, b0, b1; float fa, fb;
        map255(i, a0, a1, fa);
        map255(j, b0, b1, fb);
        float hi0v = bilerp(sHI0, PH, a0 - ohr, a1 - ohr,
                            b0 - ohc, b1 - ohc, fa, fb);

        size_t o = obase + ((size_t)i << 8) + (size_t)j;
        float highv = (hi0v + hi1v) * 0.5f;   // sum of 2 level contributions / level
        __builtin_nontemporal_store(lowv,  &out_low[o]);
        __builtin_nontemporal_store(highv, &out_high[o]);
    }
}

extern "C" void kernel_launch(void* const* d_in, const int* in_sizes, int n_in,
                              void* d_out, int out_size, void* d_ws, size_t ws_size,
                              hipStream_t stream) {
    (void)n_in; (void)d_ws; (void)ws_size; (void)out_size;
    const float* x = (const float*)d_in[0];
    // d_in[1] holds `level` (== 2 in this workload); kernel is specialized for
    // level == 2 on 256x256 planes per the reference setup.
    const int n    = in_sizes[0];            // B*C*256*256
    const int nimg = n / (256 * 256);
    float* out_low  = (float*)d_out;
    float* out_high = out_low + (size_t)n;   // outputs concatenated in return order

    dim3 grid(256 / TILE, 256 / TILE, nimg); // (4, 4, 512)
    dim3 block(NT);
    haar2_fused_kernel<<<grid, block, 0, stream>>>(x, out_low, out_high);
}